// PGModel_67542655696994
// MI455X (gfx1250) — compile-verified
//
#include <hip/hip_runtime.h>

typedef float v2f __attribute__((ext_vector_type(2)));
typedef float v8f __attribute__((ext_vector_type(8)));

#define IN_DIM    256
#define HID       128
#define NUM_CLASS 40
#define FEAT      (2 * (HID + NUM_CLASS))   // 336

// ---------------------------------------------------------------------------
// Kernel 1: x_trans = relu(x @ W1 + b1), fp32 via V_WMMA_F32_16X16X4_F32.
// Block = 256 threads = 8 waves. Each wave computes one 16x16 tile of the
// output; the block covers 16 rows x 128 (all) columns. K=256 -> 64 WMMAs.
//
// A (16x4 fp32, 2 VGPRs): lane<16 holds row M=lane, K={k0,k0+1};
//                         lane>=16 holds row M=lane-16, K={k0+2,k0+3}.
// B (4x16 fp32, 2 VGPRs): VGPR j, lane-half h, lane&15=n ->
//                         element (K=k0+j+2h, N=N0+n).
// C/D (16x16 f32, 8 VGPRs): VGPR r, lane-half h -> M = r + 8h, N = lane&15.
// ---------------------------------------------------------------------------
__global__ __launch_bounds__(256) void node_mlp_wmma(
    const float* __restrict__ x, const float* __restrict__ W1,
    const float* __restrict__ b1, float* __restrict__ xtr, int Nn)
{
    const int lane = threadIdx.x & 31;
    const int wave = threadIdx.x >> 5;        // 0..7 -> N tile
    const int M0   = blockIdx.x * 16;
    const int N0   = wave * 16;
    const int half = lane >> 4;               // 0 or 1
    const int l16  = lane & 15;

    int mrow = M0 + l16;
    if (mrow >= Nn) mrow = Nn - 1;            // clamp loads; EXEC stays all-1s

    const float* arow = x  + (size_t)mrow * IN_DIM + 2 * half;
    const float* bcol = W1 + (size_t)(2 * half) * HID + N0 + l16;

    v8f acc = {};
#pragma unroll 8
    for (int k0 = 0; k0 < IN_DIM; k0 += 4) {
        v2f a = *(const v2f*)(arow + k0);
        v2f b;
        b.x = bcol[(size_t)(k0 + 0) * HID];
        b.y = bcol[(size_t)(k0 + 1) * HID];
        acc = __builtin_amdgcn_wmma_f32_16x16x4_f32(
            /*neg_a=*/false, a, /*neg_b=*/false, b,
            /*c_mod=*/(short)0, acc, /*reuse_a=*/false, /*reuse_b=*/false);
    }

    const float bias = b1[N0 + l16];
    float* orow = xtr + (size_t)(M0 + 8 * half) * HID + N0 + l16;

    if (M0 + 16 <= Nn) {
        // Fast path (always taken when Nn % 16 == 0): 8 unguarded stores.
#pragma unroll
        for (int r = 0; r < 8; ++r) {
            float v = acc[r] + bias;
            v = v > 0.f ? v : 0.f;
            orow[(size_t)r * HID] = v;
        }
    } else {
        const int rowbase = M0 + 8 * half;
#pragma unroll
        for (int r = 0; r < 8; ++r) {
            if (rowbase + r < Nn) {
                float v = acc[r] + bias;
                v = v > 0.f ? v : 0.f;
                orow[(size_t)r * HID] = v;
            }
        }
    }
}

// ---------------------------------------------------------------------------
// Kernel 2: edge_prob[e] = sigmoid(concat(xs,ys,xt,yt) . W2 + b2).
// One wave32 per edge; 8 waves (8 edges) per 256-thread block.
// W2 (336) + b2 staged in LDS once per block. Row gathers are coalesced:
// consecutive lanes read consecutive addresses of each gathered row, and the
// whole gather working set (33.6 MB) is resident in the 192 MB L2.
// ---------------------------------------------------------------------------
__global__ __launch_bounds__(256) void edge_mlp(
    const float* __restrict__ xtr, const float* __restrict__ y_prob,
    const long long* __restrict__ src, const long long* __restrict__ tgt,
    const float* __restrict__ W2, const float* __restrict__ b2,
    float* __restrict__ out, int E)
{
    __shared__ float w[FEAT + 1];
    for (int i = threadIdx.x; i < FEAT + 1; i += blockDim.x)
        w[i] = (i < FEAT) ? W2[i] : b2[0];
    __syncthreads();

    const int lane = threadIdx.x & 31;
    const int e    = blockIdx.x * 8 + (threadIdx.x >> 5);
    if (e >= E) return;

    const long long s = src[e];
    const long long t = tgt[e];
    const float* xs = xtr    + (size_t)s * HID;
    const float* xt = xtr    + (size_t)t * HID;
    const float* ys = y_prob + (size_t)s * NUM_CLASS;
    const float* yt = y_prob + (size_t)t * NUM_CLASS;

    float sum = 0.f;
#pragma unroll
    for (int j = 0; j < 4; ++j)                       // xs: feat [0,128)
        sum += xs[lane + 32 * j] * w[lane + 32 * j];
    sum += ys[lane] * w[HID + lane];                  // ys: feat [128,168)
    if (lane < 8)
        sum += ys[32 + lane] * w[HID + 32 + lane];
#pragma unroll
    for (int j = 0; j < 4; ++j)                       // xt: feat [168,296)
        sum += xt[lane + 32 * j] * w[168 + lane + 32 * j];
    sum += yt[lane] * w[296 + lane];                  // yt: feat [296,336)
    if (lane < 8)
        sum += yt[32 + lane] * w[328 + lane];

    // wave32 tree reduction
#pragma unroll
    for (int off = 16; off > 0; off >>= 1)
        sum += __shfl_xor(sum, off, 32);

    if (lane == 0) {
        const float z = sum + w[FEAT];
        out[e] = 1.0f / (1.0f + __expf(-z));
    }
}

// ---------------------------------------------------------------------------
// d_in order: x, y_prob, edge_index_train(int64, [2,E]), W1, b1, W2, b2
// d_out: [E] fp32 ; d_ws: x_trans scratch (N*HID fp32 = 25.6 MB)
// ---------------------------------------------------------------------------
extern "C" void kernel_launch(void* const* d_in, const int* in_sizes, int n_in,
                              void* d_out, int out_size, void* d_ws, size_t ws_size,
                              hipStream_t stream)
{
    const float*     x      = (const float*)d_in[0];
    const float*     y_prob = (const float*)d_in[1];
    const long long* ei     = (const long long*)d_in[2];
    const float*     W1     = (const float*)d_in[3];
    const float*     b1     = (const float*)d_in[4];
    const float*     W2     = (const float*)d_in[5];
    const float*     b2     = (const float*)d_in[6];
    float*           out    = (float*)d_out;

    const int Nn = in_sizes[0] / IN_DIM;   // 50000 nodes
    const int E  = in_sizes[2] / 2;        // 800000 edges
    float* xtr = (float*)d_ws;             // [Nn, HID]

    const int mtiles = (Nn + 15) / 16;     // 3125
    node_mlp_wmma<<<mtiles, 256, 0, stream>>>(x, W1, b1, xtr, Nn);

    const int eblocks = (E + 7) / 8;       // 8 edges per 256-thread block
    edge_mlp<<<eblocks, 256, 0, stream>>>(xtr, y_prob, ei, ei + E, W2, b2, out, E);
}